// STN_19370302505190
// MI455X (gfx1250) — compile-verified
//
#include <hip/hip_runtime.h>
#include <hip/hip_bf16.h>

// ---------------- problem constants ----------------
#define Bsz   512
#define Hin   64
#define Win   128
#define Cin   3
#define OH    13
#define OW    26
#define NPIX  (Bsz*OH*OW)          // 173056
#define FEATN 21632                // 13*26*64
#define KCH   832                  // K chunk per wave
#define KSPLIT 26                  // 26*832 = 21632

typedef float v2f __attribute__((ext_vector_type(2)));
typedef float v8f __attribute__((ext_vector_type(8)));

#if defined(__HIP_DEVICE_COMPILE__) && defined(__gfx1250__) && \
    __has_builtin(__builtin_amdgcn_wmma_f32_16x16x4_f32)
#define USE_WMMA 1
#else
#define USE_WMMA 0
#endif

// ---------------------------------------------------------------------------
// Kernel 1: fused avg-pool(3x3,SAME,count-normalized) -> conv0(3x3,s5, pad 0)
//           and conv1(5x5,s5, pad t0/b1/l1/r1), both ReLU, concat -> feat.
// Block = 256 threads = 4 pixels x 64 output channels (ch<32: conv0, else conv1).
// ---------------------------------------------------------------------------
__global__ __launch_bounds__(256) void stn_conv(const float* __restrict__ x,
                                                const float* __restrict__ w0,
                                                const float* __restrict__ w1,
                                                float* __restrict__ feat)
{
    __shared__ float xp[4][6 * 5 * 3];   // rows oh*5-1 .. oh*5+4, cols ow*5-1 .. ow*5+3
    __shared__ float pl[4][3 * 3 * 3];   // pooled values at conv0 taps

    const int g  = threadIdx.x >> 6;     // pixel group 0..3
    const int lt = threadIdx.x & 63;     // local channel thread
    const int p  = blockIdx.x * 4 + g;   // pixel id (43264*4 == NPIX exactly)
    const int b  = p / (OH * OW);
    const int r  = p - b * (OH * OW);
    const int oh = r / OW;
    const int ow = r - oh * OW;
    const int r0 = oh * 5 - 1;
    const int c0 = ow * 5 - 1;

    // stage the union input patch (zero padded OOB)
    for (int e = lt; e < 90; e += 64) {
        int rr = e / 15, q = e - rr * 15;
        int cc = q / 3,  ch = q - cc * 3;
        int ih = r0 + rr, iw = c0 + cc;
        float v = 0.0f;
        if (ih >= 0 && ih < Hin && iw >= 0 && iw < Win)
            v = x[(((size_t)b * Hin + ih) * Win + iw) * Cin + ch];
        xp[g][e] = v;
    }
    __syncthreads();

    // pooled[kh][kw][ci] for the 3x3 conv0 taps (count-normalized SAME avg pool)
    if (lt < 27) {
        int phl = lt / 9, q = lt - phl * 9;
        int pwl = q / 3,  ch = q - pwl * 3;
        float s = 0.0f;
        for (int dh = 0; dh < 3; ++dh)
            for (int dw = 0; dw < 3; ++dw)
                s += xp[g][((phl + dh) * 5 + (pwl + dw)) * 3 + ch];
        int ph = oh * 5 + phl;               // <= 62, never 63
        int pw = ow * 5 + pwl;               // can be 0 or 127
        float cr = (ph == 0) ? 2.0f : 3.0f;
        float cc = (pw == 0 || pw == Win - 1) ? 2.0f : 3.0f;
        pl[g][lt] = s / (cr * cc);
    }
    __syncthreads();

    const int co = lt & 31;
    float acc = 0.0f;
    if (lt < 32) {          // conv0 on pooled, no padding
        for (int kh = 0; kh < 3; ++kh)
            for (int kw = 0; kw < 3; ++kw)
                for (int ci = 0; ci < 3; ++ci)
                    acc += pl[g][(kh * 3 + kw) * 3 + ci] *
                           w0[(size_t)(((kh * 3 + kw) * 3 + ci)) * 32 + co];
    } else {                // conv1 on raw x; patch row = kh+1, col = kw
        for (int kh = 0; kh < 5; ++kh)
            for (int kw = 0; kw < 5; ++kw)
                for (int ci = 0; ci < 3; ++ci)
                    acc += xp[g][((kh + 1) * 5 + kw) * 3 + ci] *
                           w1[(size_t)(((kh * 5 + kw) * 3 + ci)) * 32 + co];
    }
    acc = fmaxf(acc, 0.0f);
    feat[(size_t)p * 64 + lt] = acc;     // exactly the reference flatten order
}

// ---------------------------------------------------------------------------
// Kernel 2: fc0 GEMM with V_WMMA_F32_16X16X4_F32. One wave per block.
// grid = (32 M-tiles, 26 K-chunks). Each wave owns both 16-wide N halves,
// so feat (A) is streamed once. Partials go to ws for a cheap reduction.
// A layout (doc): lanes 0-15 M=lane, VGPR0/1 = K {0,1}; lanes 16-31 = K {2,3}.
// B layout (symmetric): lanes 0-15 N=lane, VGPR0/1 = K {0,1}; hi half K {2,3}.
// C layout (doc): VGPR r, lanes 0-15 -> (M=r, N=lane); lanes 16-31 -> (M=8+r).
// ---------------------------------------------------------------------------
__global__ __launch_bounds__(32) void fc0_wmma(const float* __restrict__ feat,
                                               const float* __restrict__ w,
                                               float* __restrict__ part)
{
    const int lane = threadIdx.x;
    const int m0   = blockIdx.x * 16;
    const int kbeg = blockIdx.y * KCH;
    const int mrow = lane & 15;
    const int hi   = lane >> 4;

    v8f acc0 = {};
    v8f acc1 = {};

#if USE_WMMA
    const float* ap = feat + (size_t)(m0 + mrow) * FEATN + kbeg + 2 * hi;
    const float* bp = w + (size_t)(kbeg + 2 * hi) * 32 + mrow;
#pragma unroll 4
    for (int k = 0; k < KCH; k += 4) {
        v2f a = *(const v2f*)(ap + k);           // global_load_b64, 8B aligned
        const float* bk = bp + (size_t)k * 32;
        v2f b0, b1;
        b0.x = bk[0];   b0.y = bk[32];           // K pair, N = mrow
        b1.x = bk[16];  b1.y = bk[48];           // K pair, N = mrow + 16
        acc0 = __builtin_amdgcn_wmma_f32_16x16x4_f32(
            false, a, false, b0, (short)0, acc0, false, false);
        acc1 = __builtin_amdgcn_wmma_f32_16x16x4_f32(
            false, a, false, b1, (short)0, acc1, false, false);
    }
#else
    // Correct VALU fallback producing the same per-lane C layout.
    for (int rr = 0; rr < 8; ++rr) {
        const float* fr = feat + (size_t)(m0 + rr + 8 * hi) * FEATN + kbeg;
        float s0 = 0.0f, s1 = 0.0f;
        for (int k = 0; k < KCH; ++k) {
            float fv = fr[k];
            s0 += fv * w[(size_t)(kbeg + k) * 32 + mrow];
            s1 += fv * w[(size_t)(kbeg + k) * 32 + mrow + 16];
        }
        acc0[rr] = s0;
        acc1[rr] = s1;
    }
#endif

    float* pp = part + (size_t)blockIdx.y * (512 * 32)
                     + (size_t)(m0 + 8 * hi) * 32 + mrow;
#pragma unroll
    for (int rr = 0; rr < 8; ++rr) {
        pp[rr * 32]      = acc0[rr];
        pp[rr * 32 + 16] = acc1[rr];
    }
}

// ---------------------------------------------------------------------------
// Kernel 3: reduce K-split partials + bias + tanh -> hid [512, 32]
// ---------------------------------------------------------------------------
__global__ __launch_bounds__(32) void fc0_finish(const float* __restrict__ part,
                                                 const float* __restrict__ bias,
                                                 float* __restrict__ hid)
{
    const int b = blockIdx.x;
    const int n = threadIdx.x;
    float s = bias[n];
    for (int c = 0; c < KSPLIT; ++c)
        s += part[(size_t)c * (512 * 32) + b * 32 + n];
    hid[b * 32 + n] = tanhf(s);
}

// ---------------------------------------------------------------------------
// Kernel 4: fc1 + tanh -> theta [512, 6]
// ---------------------------------------------------------------------------
__global__ __launch_bounds__(256) void fc1_theta(const float* __restrict__ hid,
                                                 const float* __restrict__ w,
                                                 const float* __restrict__ bias,
                                                 float* __restrict__ theta)
{
    const int i = blockIdx.x * blockDim.x + threadIdx.x;
    if (i >= 512 * 6) return;
    const int b = i / 6, j = i - b * 6;
    float s = bias[j];
    for (int k = 0; k < 32; ++k)
        s += hid[b * 32 + k] * w[k * 6 + j];
    theta[i] = tanhf(s);
}

// ---------------------------------------------------------------------------
// Kernel 5: affine grid + bilinear sampling. 1 thread per (b,h,w) pixel.
// x (50 MB) lives in L2, so the gathers are cheap; writes are coalesced.
// ---------------------------------------------------------------------------
__global__ __launch_bounds__(256) void stn_sample(const float* __restrict__ x,
                                                  const float* __restrict__ theta,
                                                  float* __restrict__ out)
{
    const int idx = blockIdx.x * 256 + threadIdx.x;    // 4194304 exact
    const int b   = idx >> 13;                          // / (64*128)
    const int rem = idx & 8191;
    const int h   = rem >> 7;
    const int wq  = rem & 127;

    const float* t = theta + b * 6;
    const float xt = -1.0f + (float)wq * (2.0f / 127.0f);
    const float yt = -1.0f + (float)h  * (2.0f / 63.0f);
    const float Tx = t[0] * xt + t[1] * yt + t[2];
    const float Ty = t[3] * xt + t[4] * yt + t[5];
    const float gx = (Tx + 1.0f) * 64.0f;               // * W/2
    const float gy = (Ty + 1.0f) * 32.0f;               // * H/2

    const float fx  = floorf(gx), fy = floorf(gy);
    const float x0f = fminf(fmaxf(fx, 0.0f), 126.0f);
    const float x1f = fminf(fmaxf(fx + 1.0f, 1.0f), 127.0f);
    const float y0f = fminf(fmaxf(fy, 0.0f), 62.0f);
    const float y1f = fminf(fmaxf(fy + 1.0f, 1.0f), 63.0f);
    const int xi0 = (int)x0f, xi1 = (int)x1f;
    const int yi0 = (int)y0f, yi1 = (int)y1f;
    const float wx0 = x1f - gx, wx1 = gx - x0f;
    const float wy0 = y1f - gy, wy1 = gy - y0f;
    const float inv = 1.0f / ((x1f - x0f) * (y1f - y0f));

    const float* xb  = x + (size_t)b * (Hin * Win * Cin);
    const float* p00 = xb + (yi0 * Win + xi0) * 3;
    const float* p01 = xb + (yi1 * Win + xi0) * 3;
    const float* p10 = xb + (yi0 * Win + xi1) * 3;
    const float* p11 = xb + (yi1 * Win + xi1) * 3;
    float* o = out + (size_t)idx * 3;
#pragma unroll
    for (int c = 0; c < 3; ++c) {
        float v = wx0 * (wy0 * p00[c] + wy1 * p01[c]) +
                  wx1 * (wy0 * p10[c] + wy1 * p11[c]);
        o[c] = v * inv;
    }
}

// ---------------------------------------------------------------------------
extern "C" void kernel_launch(void* const* d_in, const int* in_sizes, int n_in,
                              void* d_out, int out_size, void* d_ws, size_t ws_size,
                              hipStream_t stream) {
    (void)in_sizes; (void)n_in; (void)out_size; (void)ws_size;
    const float* x   = (const float*)d_in[0];
    const float* w0  = (const float*)d_in[1];   // [3,3,3,32] HWIO
    const float* w1  = (const float*)d_in[2];   // [5,5,3,32] HWIO
    const float* fw0 = (const float*)d_in[3];   // [21632,32]
    const float* fb0 = (const float*)d_in[4];   // [32]
    const float* fw1 = (const float*)d_in[5];   // [32,6]
    const float* fb1 = (const float*)d_in[6];   // [6]

    char* ws = (char*)d_ws;
    float* feat  = (float*)(ws);                       // 512*21632*4 = 44302336 B
    float* part  = (float*)(ws + 44302336);            // 26*512*32*4 =  1703936 B
    float* hid   = (float*)(ws + 46006272);            // 512*32*4    =    65536 B
    float* theta = (float*)(ws + 46071808);            // 512*6*4     =    12288 B

    stn_conv  <<<NPIX / 4, 256, 0, stream>>>(x, w0, w1, feat);
    fc0_wmma  <<<dim3(32, KSPLIT), 32, 0, stream>>>(feat, fw0, part);
    fc0_finish<<<512, 32, 0, stream>>>(part, fb0, hid);
    fc1_theta <<<12, 256, 0, stream>>>(hid, fw1, fb1, theta);
    stn_sample<<<(Bsz * Hin * Win) / 256, 256, 0, stream>>>(x, theta, (float*)d_out);
}